// HOMConv_31147102831210
// MI455X (gfx1250) — compile-verified
//
#include <hip/hip_runtime.h>

typedef __attribute__((ext_vector_type(2))) float v2f;
typedef __attribute__((ext_vector_type(8))) float v8f;

#define D_FEAT 128

// ---------------------------------------------------------------------------
// Zero workspace (agg, nbr, deg). Grid-stride so one launch covers everything.
// ---------------------------------------------------------------------------
__global__ void zero_f32_kernel(float* __restrict__ p, size_t n) {
    size_t i = (size_t)blockIdx.x * blockDim.x + threadIdx.x;
    size_t stride = (size_t)gridDim.x * blockDim.x;
    for (; i < n; i += stride) p[i] = 0.0f;
}

// ---------------------------------------------------------------------------
// In-degree via f32 atomics (deg used as float downstream anyway).
// ---------------------------------------------------------------------------
__global__ void deg_kernel(const int* __restrict__ edst, float* __restrict__ deg,
                           int n_edges) {
    int e = blockIdx.x * blockDim.x + threadIdx.x;
    if (e < n_edges) atomicAdd(&deg[edst[e]], 1.0f);
}

// ---------------------------------------------------------------------------
// H = X @ W^T + b using V_WMMA_F32_16X16X4_F32 (f32 matrix path, matches
// reference precision). One wave per 16x16 output tile; 8 waves per block
// cover the 8 N-tiles (D_OUT=128) of one 16-row M block.
//
// A (16x4, f32, 2 VGPR):  lane%16 = M row; lanes 0-15 hold K=kb+0/kb+1,
//                         lanes 16-31 hold K=kb+2/kb+3.
// B (4x16, f32, 2 VGPR):  lane%16 = N col; same K split across lane halves.
// C/D (16x16 f32, 8 VGPR): VGPR g -> row g (lanes 0-15) / g+8 (lanes 16-31),
//                          col = lane%16.
// EXEC must be all-ones for WMMA: reads are clamped, only stores are guarded.
// ---------------------------------------------------------------------------
__global__ __launch_bounds__(256) void gemm_wmma_kernel(
        const float* __restrict__ X, const float* __restrict__ W,
        const float* __restrict__ bias, float* __restrict__ H, int n_nodes) {
    const int lane  = threadIdx.x & 31;
    const int wave  = threadIdx.x >> 5;      // 0..7 -> N tile
    const int tileM = blockIdx.x * 16;
    const int tileN = wave * 16;
    const int col   = lane & 15;             // M row for A, N col for B/C/D
    const int hi    = lane >> 4;             // K half selector

    int arow = tileM + col;
    if (arow > n_nodes - 1) arow = n_nodes - 1;   // clamp, keep EXEC full

    const float* Ap = X + (size_t)arow * D_FEAT + 2 * hi;           // A[m][k]
    const float* Bp = W + (size_t)(tileN + col) * D_FEAT + 2 * hi;  // B[k][n]=W[n][k]

    v8f c = {};
#pragma unroll
    for (int k = 0; k < D_FEAT; k += 4) {
        v2f a = *(const v2f*)(Ap + k);
        v2f b = *(const v2f*)(Bp + k);
        c = __builtin_amdgcn_wmma_f32_16x16x4_f32(
                /*neg_a=*/false, a, /*neg_b=*/false, b,
                /*c_mod=*/(short)0, c, /*reuse_a=*/false, /*reuse_b=*/false);
    }

    const float bb = bias[tileN + col];
    float* Hout = H + (size_t)tileM * D_FEAT + tileN + col;
#pragma unroll
    for (int g = 0; g < 8; ++g) {
        int m = g + 8 * hi;
        if (tileM + m < n_nodes) Hout[(size_t)m * D_FEAT] = c[g] + bb;
    }
}

// ---------------------------------------------------------------------------
// Edge scatter: one wave per edge, lane j owns feature quad [4j..4j+3].
// Gathers H[src] (L2-resident: H=51MB) as float4, scatter-adds both the
// src-normalized (agg) and raw (nbr) messages with f32 global atomics
// (accumulators 102MB, also L2-resident on the 192MB L2).
// ---------------------------------------------------------------------------
__global__ __launch_bounds__(256) void edge_scatter_kernel(
        const float* __restrict__ H, const float* __restrict__ deg,
        const int* __restrict__ esrc, const int* __restrict__ edst,
        float* __restrict__ agg, float* __restrict__ nbr, int n_edges) {
    long long t = (long long)blockIdx.x * blockDim.x + threadIdx.x;
    int e = (int)(t >> 5);
    if (e >= n_edges) return;
    int q = ((int)t & 31) * 4;

    int s = esrc[e];
    int d = edst[e];
    float w = rsqrtf(deg[s] + 1.0f);

    const float4 h = *(const float4*)(H + (size_t)s * D_FEAT + q);
    float* pa = agg + (size_t)d * D_FEAT + q;
    float* pn = nbr + (size_t)d * D_FEAT + q;

    atomicAdd(pa + 0, h.x * w);
    atomicAdd(pa + 1, h.y * w);
    atomicAdd(pa + 2, h.z * w);
    atomicAdd(pa + 3, h.w * w);
    atomicAdd(pn + 0, h.x);
    atomicAdd(pn + 1, h.y);
    atomicAdd(pn + 2, h.z);
    atomicAdd(pn + 3, h.w);
}

// ---------------------------------------------------------------------------
// out = relu(0.5 * (inv_sqrt*agg + inv_sqrt^2*H + nbr/max(deg,1)))
// Elementwise, in-place over H (which lives in d_out). float4 per thread.
// ---------------------------------------------------------------------------
__global__ __launch_bounds__(256) void finalize_kernel(
        float* __restrict__ H, const float* __restrict__ agg,
        const float* __restrict__ nbr, const float* __restrict__ deg,
        int n_nodes) {
    long long t = (long long)blockIdx.x * blockDim.x + threadIdx.x;
    int v = (int)(t >> 5);
    if (v >= n_nodes) return;
    int q = ((int)t & 31) * 4;

    float dg   = deg[v];
    float is   = rsqrtf(dg + 1.0f);
    float is2  = is * is;
    float invd = 1.0f / fmaxf(dg, 1.0f);

    size_t base = (size_t)v * D_FEAT + q;
    float4 h = *(float4*)(H + base);
    float4 a = *(const float4*)(agg + base);
    float4 n = *(const float4*)(nbr + base);

    float4 o;
    o.x = fmaxf(0.0f, 0.5f * (is * a.x + is2 * h.x + n.x * invd));
    o.y = fmaxf(0.0f, 0.5f * (is * a.y + is2 * h.y + n.y * invd));
    o.z = fmaxf(0.0f, 0.5f * (is * a.z + is2 * h.z + n.z * invd));
    o.w = fmaxf(0.0f, 0.5f * (is * a.w + is2 * h.w + n.w * invd));
    *(float4*)(H + base) = o;
}

// ---------------------------------------------------------------------------
extern "C" void kernel_launch(void* const* d_in, const int* in_sizes, int n_in,
                              void* d_out, int out_size, void* d_ws, size_t ws_size,
                              hipStream_t stream) {
    const float* X    = (const float*)d_in[0];
    const float* W    = (const float*)d_in[1];
    const float* bias = (const float*)d_in[2];
    const int* esrc   = (const int*)d_in[3];
    const int* edst   = (const int*)d_in[4];

    const int n_nodes = in_sizes[0] / D_FEAT;
    const int n_edges = in_sizes[3];

    // H lives in d_out (finalize is elementwise in-place safe).
    float* H   = (float*)d_out;
    float* agg = (float*)d_ws;
    float* nbr = agg + (size_t)n_nodes * D_FEAT;
    float* deg = nbr + (size_t)n_nodes * D_FEAT;

    // 1. Zero accumulators + degree (d_ws is poisoned by harness).
    size_t nz = (size_t)2 * n_nodes * D_FEAT + n_nodes;
    zero_f32_kernel<<<4096, 256, 0, stream>>>(agg, nz);

    // 2. In-degree.
    deg_kernel<<<(n_edges + 255) / 256, 256, 0, stream>>>(edst, deg, n_edges);

    // 3. H = X @ W^T + b  (WMMA f32 16x16x4; 8 waves/block = all 8 N tiles).
    int m_tiles = (n_nodes + 15) / 16;
    gemm_wmma_kernel<<<m_tiles, 256, 0, stream>>>(X, W, bias, H, n_nodes);

    // 4. Edge gather + dual scatter-add (one wave per edge).
    long long sc_threads = (long long)n_edges * 32;
    int sc_blocks = (int)((sc_threads + 255) / 256);
    edge_scatter_kernel<<<sc_blocks, 256, 0, stream>>>(H, deg, esrc, edst,
                                                       agg, nbr, n_edges);

    // 5. Fuse spectral + spatial branches, ReLU, in place over d_out.
    long long fn_threads = (long long)n_nodes * 32;
    int fn_blocks = (int)((fn_threads + 255) / 256);
    finalize_kernel<<<fn_blocks, 256, 0, stream>>>(H, agg, nbr, deg, n_nodes);
}